// EnhancedTelomeraseGNN_74646531604976
// MI455X (gfx1250) — compile-verified
//
#include <hip/hip_runtime.h>
#include <math.h>

#define NNODE 50000
#define NEDGE 400000
#define F_IN  32
#define HID   64
#define HEADS 4
#define NT    20
#define NB    64
#define IN1   96   // F_IN + HID

typedef __attribute__((ext_vector_type(2))) float v2f;
typedef __attribute__((ext_vector_type(8))) float v8f;

#define CDIV(a,b) (((a)+(b)-1)/(b))

__device__ __forceinline__ float eluf(float x)   { return x > 0.0f ? x : expf(x) - 1.0f; }
__device__ __forceinline__ float lrelu(float x)  { return x > 0.0f ? x : 0.2f * x; }

// monotone float<->uint encoding for atomic float max
__device__ __forceinline__ unsigned fenc(float f) {
    unsigned u = __float_as_uint(f);
    return (u & 0x80000000u) ? ~u : (u | 0x80000000u);
}
__device__ __forceinline__ float fdec(unsigned u) {
    unsigned b = (u & 0x80000000u) ? (u & 0x7FFFFFFFu) : ~u;
    return __uint_as_float(b);
}
#define NEG_INF_ENC 0x007FFFFFu   // fenc(-inf)

// ---------------------------------------------------------------- fill
__global__ void fill_u32(unsigned* __restrict__ p, size_t n, unsigned v) {
    size_t i = (size_t)blockIdx.x * blockDim.x + threadIdx.x;
    if (i < n) p[i] = v;
}

// ---------------------------------------------------------------- edge-attention coefficients
// coef[0..3]=ke1, [4..7]=ce1, [8..11]=ke2, [12..15]=ce2
__global__ void coef_kernel(const float* __restrict__ edge_W, const float* __restrict__ edge_b,
                            const float* __restrict__ We1, const float* __restrict__ att_e1,
                            const float* __restrict__ We2, const float* __restrict__ att_e2,
                            float* __restrict__ coef) {
    __shared__ float red[256];
    int j = threadIdx.x;          // j = h*64 + c
    int grp = j >> 6;
    for (int pass = 0; pass < 4; ++pass) {
        const float* vec = (pass & 1) ? edge_b : edge_W;
        const float* We  = (pass < 2) ? We1 : We2;
        const float* att = (pass < 2) ? att_e1 : att_e2;
        float proj = 0.0f;
        for (int k = 0; k < HID; ++k) proj += vec[k] * We[k * 256 + j];
        red[j] = proj * att[j];
        __syncthreads();
        for (int s = 32; s > 0; s >>= 1) {
            if ((j & 63) < s) red[j] += red[j + s];
            __syncthreads();
        }
        if ((j & 63) == 0) coef[pass * 4 + grp] = red[j];
        __syncthreads();
    }
}

// ---------------------------------------------------------------- concat x with node-type embedding
__global__ void concat_kernel(const float* __restrict__ x, const int* __restrict__ node_type,
                              const float* __restrict__ node_emb, float* __restrict__ h0) {
    int idx = blockIdx.x * blockDim.x + threadIdx.x;
    if (idx >= NNODE * IN1) return;
    int n = idx / IN1, c = idx % IN1;
    h0[idx] = (c < F_IN) ? x[n * F_IN + c]
                         : node_emb[node_type[n] * HID + (c - F_IN)];
}

// ---------------------------------------------------------------- f32 WMMA GEMM: C[M,Nc] = A[M,K] @ B[K,Nc] (+C)
// one wave per 16x16 output tile; M%16==0, K%4==0, Nc%16==0
__global__ void wmma_gemm_f32(const float* __restrict__ A, const float* __restrict__ B,
                              float* __restrict__ C, int M, int K, int Nc, int accumulate) {
    int wave = (blockIdx.x * blockDim.x + threadIdx.x) >> 5;
    int lane = threadIdx.x & 31;
    int ntn = Nc >> 4;
    int total = (M >> 4) * ntn;
    if (wave >= total) return;                 // wave-uniform exit: EXEC stays all-1s
    int m0 = (wave / ntn) << 4;
    int n0 = (wave % ntn) << 4;
    int hi = lane >> 4;                        // half-wave select
    int lm = lane & 15;

    v8f acc = {};
    if (accumulate) {
#pragma unroll
        for (int v = 0; v < 8; ++v) acc[v] = C[(size_t)(m0 + v + 8 * hi) * Nc + n0 + lm];
    }
    for (int k = 0; k < K; k += 4) {
        v2f a, b;
        a[0] = A[(size_t)(m0 + lm) * K + k + 2 * hi];
        a[1] = A[(size_t)(m0 + lm) * K + k + 2 * hi + 1];
        b[0] = B[(size_t)(k + 2 * hi) * Nc + n0 + lm];
        b[1] = B[(size_t)(k + 2 * hi + 1) * Nc + n0 + lm];
        acc = __builtin_amdgcn_wmma_f32_16x16x4_f32(false, a, false, b, (short)0, acc, false, false);
    }
#pragma unroll
    for (int v = 0; v < 8; ++v) C[(size_t)(m0 + v + 8 * hi) * Nc + n0 + lm] = acc[v];
}

// ---------------------------------------------------------------- per-node attention logits
__global__ void attn_kernel(const float* __restrict__ h1, const float* __restrict__ att_s,
                            const float* __restrict__ att_d, float* __restrict__ a_s,
                            float* __restrict__ a_d) {
    int idx = blockIdx.x * blockDim.x + threadIdx.x;   // n*4 + h
    if (idx >= NNODE * HEADS) return;
    int n = idx >> 2, h = idx & 3;
    const float* hp = h1 + (size_t)n * 256 + h * 64;
    float s1 = 0.0f, s2 = 0.0f;
    for (int c = 0; c < HID; ++c) {
        float v = hp[c];
        s1 += v * att_s[h * 64 + c];
        s2 += v * att_d[h * 64 + c];
    }
    a_s[idx] = s1;
    a_d[idx] = s2;
}

// ---------------------------------------------------------------- segment max of alpha over dst
__global__ void edge_max_kernel(const int* __restrict__ src, const int* __restrict__ dst,
                                const float* __restrict__ edge_attr,
                                const float* __restrict__ a_s, const float* __restrict__ a_d,
                                const float* __restrict__ ke, const float* __restrict__ ce,
                                unsigned* __restrict__ mEnc) {
    int idx = blockIdx.x * blockDim.x + threadIdx.x;   // e*4 + h
    if (idx >= NEDGE * HEADS) return;
    int e = idx >> 2, h = idx & 3;
    int s = src[e], d = dst[e];
    float alpha = lrelu(a_s[s * 4 + h] + a_d[d * 4 + h] + ke[h] * edge_attr[e] + ce[h]);
    atomicMax(&mEnc[d * 4 + h], fenc(alpha));
}

// ---------------------------------------------------------------- exp(alpha - m), denominator
__global__ void edge_exp_kernel(const int* __restrict__ src, const int* __restrict__ dst,
                                const float* __restrict__ edge_attr,
                                const float* __restrict__ a_s, const float* __restrict__ a_d,
                                const float* __restrict__ ke, const float* __restrict__ ce,
                                const unsigned* __restrict__ mEnc,
                                float* __restrict__ exb, float* __restrict__ den) {
    int idx = blockIdx.x * blockDim.x + threadIdx.x;   // e*4 + h
    if (idx >= NEDGE * HEADS) return;
    int e = idx >> 2, h = idx & 3;
    int s = src[e], d = dst[e];
    float alpha = lrelu(a_s[s * 4 + h] + a_d[d * 4 + h] + ke[h] * edge_attr[e] + ce[h]);
    float m = fdec(mEnc[d * 4 + h]);
    if (!isfinite(m)) m = 0.0f;
    float ex = expf(alpha - m);
    exb[idx] = ex;
    atomicAdd(&den[d * 4 + h], ex);
}

// ---------------------------------------------------------------- accum[dst] += ex * h1[src]
__global__ void edge_agg_kernel(const int* __restrict__ src, const int* __restrict__ dst,
                                const float* __restrict__ exb, const float* __restrict__ h1,
                                float* __restrict__ accum) {
    int idx = blockIdx.x * blockDim.x + threadIdx.x;   // e*64 + c
    if (idx >= NEDGE * HID) return;
    int e = idx >> 6, c = idx & 63;
    int s = src[e], d = dst[e];
#pragma unroll
    for (int h = 0; h < HEADS; ++h) {
        float w = exb[e * 4 + h];
        atomicAdd(&accum[(size_t)d * 256 + h * 64 + c],
                  w * h1[(size_t)s * 256 + h * 64 + c]);
    }
}

// ---------------------------------------------------------------- normalize, head-mean, bias, elu
__global__ void gat_finalize_kernel(const float* __restrict__ accum, const float* __restrict__ den,
                                    const float* __restrict__ bias, float* __restrict__ hout) {
    int idx = blockIdx.x * blockDim.x + threadIdx.x;   // n*64 + c
    if (idx >= NNODE * HID) return;
    int n = idx >> 6, c = idx & 63;
    float s = 0.0f;
#pragma unroll
    for (int h = 0; h < HEADS; ++h)
        s += accum[(size_t)n * 256 + h * 64 + c] / (den[n * 4 + h] + 1e-16f);
    hout[idx] = eluf(s * 0.25f + bias[c]);
}

// ---------------------------------------------------------------- SAGE: mean-aggregate neighbors
__global__ void sage_agg_kernel(const int* __restrict__ src, const int* __restrict__ dst,
                                const float* __restrict__ h, float* __restrict__ agg,
                                float* __restrict__ deg) {
    int idx = blockIdx.x * blockDim.x + threadIdx.x;   // e*64 + c
    if (idx >= NEDGE * HID) return;
    int e = idx >> 6, c = idx & 63;
    int s = src[e], d = dst[e];
    atomicAdd(&agg[(size_t)d * HID + c], h[(size_t)s * HID + c]);
    if (c == 0) atomicAdd(&deg[d], 1.0f);
}

__global__ void sage_mean_kernel(const float* __restrict__ agg, const float* __restrict__ deg,
                                 float* __restrict__ aggm) {
    int idx = blockIdx.x * blockDim.x + threadIdx.x;
    if (idx >= NNODE * HID) return;
    int n = idx >> 6;
    aggm[idx] = agg[idx] / fmaxf(deg[n], 1.0f);
}

__global__ void sage_finalize_kernel(const float* __restrict__ tmp, const float* __restrict__ bias,
                                     float* __restrict__ h) {
    int idx = blockIdx.x * blockDim.x + threadIdx.x;
    if (idx >= NNODE * HID) return;
    h[idx] = eluf(tmp[idx] + bias[idx & 63]);
}

// ---------------------------------------------------------------- global mean pool
__global__ void pool_kernel(const int* __restrict__ batch, const float* __restrict__ h,
                            float* __restrict__ gsum, float* __restrict__ gcnt) {
    int idx = blockIdx.x * blockDim.x + threadIdx.x;   // n*64 + c
    if (idx >= NNODE * HID) return;
    int n = idx >> 6, c = idx & 63;
    int g = batch[n];
    atomicAdd(&gsum[g * HID + c], h[idx]);
    if (c == 0) atomicAdd(&gcnt[g], 1.0f);
}

// ---------------------------------------------------------------- MLP head, one block per graph
__global__ void head_kernel(const float* __restrict__ gsum, const float* __restrict__ gcnt,
                            const float* __restrict__ lin1_W, const float* __restrict__ lin1_b,
                            const float* __restrict__ lin2_W, const float* __restrict__ lin2_b,
                            const float* __restrict__ tel_W, const float* __restrict__ tel_b,
                            const float* __restrict__ comp_W, const float* __restrict__ comp_b,
                            const float* __restrict__ pch_W, const float* __restrict__ pch_b,
                            float* __restrict__ out) {
    __shared__ float sg[64], s1[64], s2[64];
    int b = blockIdx.x, j = threadIdx.x;
    sg[j] = gsum[b * 64 + j] / fmaxf(gcnt[b], 1.0f);
    __syncthreads();
    float a = lin1_b[j];
    for (int k = 0; k < 64; ++k) a += sg[k] * lin1_W[k * 64 + j];
    s1[j] = eluf(a);
    __syncthreads();
    a = lin2_b[j];
    for (int k = 0; k < 64; ++k) a += s1[k] * lin2_W[k * 64 + j];
    s2[j] = a;
    __syncthreads();
    if (j == 0) { float t = tel_b[0];  for (int k = 0; k < 64; ++k) t += s2[k] * tel_W[k];  out[b] = t; }
    if (j == 1) { float t = comp_b[0]; for (int k = 0; k < 64; ++k) t += s2[k] * comp_W[k]; out[64 + b] = t; }
    if (j == 2) { float t = pch_b[0];  for (int k = 0; k < 64; ++k) t += s2[k] * pch_W[k];  out[128 + b] = t; }
}

// ================================================================ host side
struct GatWs {
    float *h1, *a_s, *a_d, *den, *exb, *accum;
    unsigned* mEnc;
    const int *src, *dst;
    const float* edge_attr;
};

static void run_gat_layer(const float* feat_in, int K, const float* W,
                          const float* att_s, const float* att_d, const float* bias,
                          const float* ke, const float* ce, const GatWs& w,
                          float* hout, hipStream_t stream) {
    // reset segment buffers
    fill_u32<<<CDIV((size_t)NNODE * 4, 256), 256, 0, stream>>>(w.mEnc, (size_t)NNODE * 4, NEG_INF_ENC);
    fill_u32<<<CDIV((size_t)NNODE * 4, 256), 256, 0, stream>>>((unsigned*)w.den, (size_t)NNODE * 4, 0u);
    fill_u32<<<CDIV((size_t)NNODE * 256, 256), 256, 0, stream>>>((unsigned*)w.accum, (size_t)NNODE * 256, 0u);
    // projection: h1 = feat_in @ W   [N,K]@[K,256]
    {
        int tiles = (NNODE / 16) * (256 / 16);
        wmma_gemm_f32<<<CDIV(tiles, 8), 256, 0, stream>>>(feat_in, W, w.h1, NNODE, K, 256, 0);
    }
    attn_kernel<<<CDIV(NNODE * HEADS, 256), 256, 0, stream>>>(w.h1, att_s, att_d, w.a_s, w.a_d);
    edge_max_kernel<<<CDIV(NEDGE * HEADS, 256), 256, 0, stream>>>(w.src, w.dst, w.edge_attr,
                                                                  w.a_s, w.a_d, ke, ce, w.mEnc);
    edge_exp_kernel<<<CDIV(NEDGE * HEADS, 256), 256, 0, stream>>>(w.src, w.dst, w.edge_attr,
                                                                  w.a_s, w.a_d, ke, ce, w.mEnc,
                                                                  w.exb, w.den);
    edge_agg_kernel<<<CDIV(NEDGE * HID, 256), 256, 0, stream>>>(w.src, w.dst, w.exb, w.h1, w.accum);
    gat_finalize_kernel<<<CDIV(NNODE * HID, 256), 256, 0, stream>>>(w.accum, w.den, bias, hout);
}

extern "C" void kernel_launch(void* const* d_in, const int* in_sizes, int n_in,
                              void* d_out, int out_size, void* d_ws, size_t ws_size,
                              hipStream_t stream) {
    const float* x         = (const float*)d_in[0];
    const int*   ei        = (const int*)d_in[1];
    const float* edge_attr = (const float*)d_in[2];
    const int*   batch     = (const int*)d_in[3];
    const int*   node_type = (const int*)d_in[4];
    const float* node_emb  = (const float*)d_in[5];
    const float* edge_W    = (const float*)d_in[6];
    const float* edge_b    = (const float*)d_in[7];
    const float* W1        = (const float*)d_in[8];
    const float* att_src1  = (const float*)d_in[9];
    const float* att_dst1  = (const float*)d_in[10];
    const float* We1       = (const float*)d_in[11];
    const float* att_e1    = (const float*)d_in[12];
    const float* b1        = (const float*)d_in[13];
    const float* W2        = (const float*)d_in[14];
    const float* att_src2  = (const float*)d_in[15];
    const float* att_dst2  = (const float*)d_in[16];
    const float* We2       = (const float*)d_in[17];
    const float* att_e2    = (const float*)d_in[18];
    const float* b2        = (const float*)d_in[19];
    const float* sage_Wl   = (const float*)d_in[20];
    const float* sage_Wr   = (const float*)d_in[21];
    const float* sage_b    = (const float*)d_in[22];
    const float* lin1_W    = (const float*)d_in[23];
    const float* lin1_b    = (const float*)d_in[24];
    const float* lin2_W    = (const float*)d_in[25];
    const float* lin2_b    = (const float*)d_in[26];
    const float* tel_W     = (const float*)d_in[27];
    const float* tel_b     = (const float*)d_in[28];
    const float* comp_W    = (const float*)d_in[29];
    const float* comp_b    = (const float*)d_in[30];
    const float* pch_W     = (const float*)d_in[31];
    const float* pch_b     = (const float*)d_in[32];
    float* out = (float*)d_out;

    const int* src = ei;
    const int* dst = ei + NEDGE;

    // workspace layout (floats)
    float* ws = (float*)d_ws;
    size_t off = 0;
    float* h0    = ws + off; off += (size_t)NNODE * IN1;
    float* h1    = ws + off; off += (size_t)NNODE * 256;
    float* a_s   = ws + off; off += (size_t)NNODE * 4;
    float* a_d   = ws + off; off += (size_t)NNODE * 4;
    unsigned* mE = (unsigned*)(ws + off); off += (size_t)NNODE * 4;
    float* den   = ws + off; off += (size_t)NNODE * 4;
    float* exb   = ws + off; off += (size_t)NEDGE * 4;
    float* accum = ws + off; off += (size_t)NNODE * 256;
    float* hcur  = ws + off; off += (size_t)NNODE * HID;
    float* tmp   = ws + off; off += (size_t)NNODE * HID;
    float* agg   = ws + off; off += (size_t)NNODE * HID;
    float* aggm  = ws + off; off += (size_t)NNODE * HID;
    float* deg   = ws + off; off += (size_t)NNODE;
    float* gsum  = ws + off; off += (size_t)NB * HID;
    float* gcnt  = ws + off; off += (size_t)NB;
    float* coef  = ws + off; off += 16;
    (void)ws_size; (void)in_sizes; (void)n_in; (void)out_size;

    // edge-attention coefficients (collapses the rank-1 edge GEMMs)
    coef_kernel<<<1, 256, 0, stream>>>(edge_W, edge_b, We1, att_e1, We2, att_e2, coef);

    // input features: concat(x, node_emb[node_type])
    concat_kernel<<<CDIV(NNODE * IN1, 256), 256, 0, stream>>>(x, node_type, node_emb, h0);

    GatWs w;
    w.h1 = h1; w.a_s = a_s; w.a_d = a_d; w.den = den; w.exb = exb; w.accum = accum;
    w.mEnc = mE; w.src = src; w.dst = dst; w.edge_attr = edge_attr;

    // GAT layer 1:  [N,96] -> [N,64]
    run_gat_layer(h0, IN1, W1, att_src1, att_dst1, b1, coef + 0, coef + 4, w, hcur, stream);
    // GAT layer 2:  [N,64] -> [N,64]
    run_gat_layer(hcur, HID, W2, att_src2, att_dst2, b2, coef + 8, coef + 12, w, hcur, stream);

    // SAGEConv (mean aggregation)
    fill_u32<<<CDIV((size_t)NNODE, 256), 256, 0, stream>>>((unsigned*)deg, (size_t)NNODE, 0u);
    fill_u32<<<CDIV((size_t)NNODE * HID, 256), 256, 0, stream>>>((unsigned*)agg, (size_t)NNODE * HID, 0u);
    sage_agg_kernel<<<CDIV(NEDGE * HID, 256), 256, 0, stream>>>(src, dst, hcur, agg, deg);
    sage_mean_kernel<<<CDIV(NNODE * HID, 256), 256, 0, stream>>>(agg, deg, aggm);
    {
        int tiles = (NNODE / 16) * (HID / 16);
        wmma_gemm_f32<<<CDIV(tiles, 8), 256, 0, stream>>>(aggm, sage_Wl, tmp, NNODE, HID, HID, 0);
        wmma_gemm_f32<<<CDIV(tiles, 8), 256, 0, stream>>>(hcur, sage_Wr, tmp, NNODE, HID, HID, 1);
    }
    sage_finalize_kernel<<<CDIV(NNODE * HID, 256), 256, 0, stream>>>(tmp, sage_b, hcur);

    // global mean pool + MLP heads
    fill_u32<<<CDIV((size_t)(NB * HID + NB), 256), 256, 0, stream>>>((unsigned*)gsum,
                                                                     (size_t)NB * HID + NB, 0u);
    pool_kernel<<<CDIV(NNODE * HID, 256), 256, 0, stream>>>(batch, hcur, gsum, gcnt);
    head_kernel<<<NB, 64, 0, stream>>>(gsum, gcnt, lin1_W, lin1_b, lin2_W, lin2_b,
                                       tel_W, tel_b, comp_W, comp_b, pch_W, pch_b, out);
}